// Transitions_10831907521129
// MI455X (gfx1250) — compile-verified
//
#include <hip/hip_runtime.h>

// ---------------------------------------------------------------------------
// Octree "Transitions" block for MI455X (gfx1250, wave32, WMMA)
//   out0 [N6,96]  = data0 + up( relu(bn( data1 @ w10 )) )
//   out1 [N5,192] = relu(bn( pool(data0) @ w01 )) + data1
//   out2 [N4,384] = relu(bn( pool2(data0) @ w02 )) + relu(bn( pool(data1) @ w12 ))
// BN is training-mode batch stats -> two-pass: GEMM to ws, column reduce,
// fold coefficients, fused normalize+relu+residual.
// GEMMs run on V_WMMA_F32_16X16X4_F32 with compile-time shapes and weights
// pre-packed into B-fragment order (one coalesced b64 load per WMMA step).
// ---------------------------------------------------------------------------

#define N6 524288
#define N5 65536
#define N4 8192
#define BN_EPS 1e-5f

typedef __attribute__((ext_vector_type(2))) float v2f;
typedef __attribute__((ext_vector_type(8))) float v8f;

// ---------------- max-pool over 8 contiguous children (float4 vectorized) ---
__global__ void pool8_kernel(const float* __restrict__ in, float* __restrict__ out,
                             int C4, int nvec) {
    int idx = blockIdx.x * blockDim.x + threadIdx.x;
    if (idx >= nvec) return;
    int j = idx / C4;
    int c = (idx - j * C4) * 4;
    int C = C4 * 4;
    const float* p = in + (size_t)j * 8 * C + c;
    float4 m = *(const float4*)p;
#pragma unroll
    for (int r = 1; r < 8; ++r) {
        float4 v = *(const float4*)(p + (size_t)r * C);
        m.x = fmaxf(m.x, v.x); m.y = fmaxf(m.y, v.y);
        m.z = fmaxf(m.z, v.z); m.w = fmaxf(m.w, v.w);
    }
    *(float4*)(out + (size_t)j * C + c) = m;
}

// ---------------- pack W[KxN] into WMMA B-fragment order --------------------
// Wp[((nt*(K/4) + k4)*32 + lane)*2 + {0,1}] = W[k4*4 + 2h + {0,1}][nt*16 + lm]
// where lm = lane&15, h = lane>>4.  One wave then reads one contiguous 256B
// chunk per WMMA step.
__global__ void pack_w_kernel(const float* __restrict__ W, float* __restrict__ Wp,
                              int N, int K) {
    int idx = blockIdx.x * blockDim.x + threadIdx.x;        // (nt, k4, lane)
    int total = (N >> 4) * (K >> 2) * 32;
    if (idx >= total) return;
    int lane = idx & 31;
    int t    = idx >> 5;
    int K4   = K >> 2;
    int k4   = t % K4;
    int nt   = t / K4;
    int lm = lane & 15, h = lane >> 4;
    int k = k4 * 4 + 2 * h;
    int n = nt * 16 + lm;
    Wp[(size_t)idx * 2 + 0] = W[(size_t)k * N + n];
    Wp[(size_t)idx * 2 + 1] = W[(size_t)(k + 1) * N + n];
}

// ---------------- fp32 GEMM via V_WMMA_F32_16X16X4_F32 ----------------------
// One wave computes one 16x16 tile of Y = X[MxK] @ W[KxN] (W pre-packed).
// A-frag (16x4 f32): lane (h,lm) holds {X[m0+lm][k+2h], X[m0+lm][k+2h+1]}
// B-frag (4x16 f32): contiguous packed v2f per lane.
// C/D (16x16 f32):   acc[v] = Y[m0 + v + 8h][n0 + lm]
template <int N, int K>
__global__ void gemm_wmma_f32(const float* __restrict__ X, const float* __restrict__ Wp,
                              float* __restrict__ Y, int M) {
    constexpr int NT = N / 16;
    constexpr int K4 = K / 4;

    const int wid  = threadIdx.x >> 5;
    const int lane = threadIdx.x & 31;
    const int lm   = lane & 15;
    const int kh   = lane >> 4;

    int tile = blockIdx.x * (blockDim.x >> 5) + wid;
    int mt = tile / NT;
    int nt = tile - mt * NT;
    if (mt >= (M >> 4)) return;
    const int m0 = mt << 4;
    const int n0 = nt << 4;

    const float* xrow = X + (size_t)(m0 + lm) * K + 2 * kh;       // +16B per k4
    const v2f*   wfr  = (const v2f*)Wp + (size_t)nt * K4 * 32 + lane; // +256B per k4

    v8f acc = {};
#pragma unroll
    for (int k4 = 0; k4 < K4; ++k4) {
        v2f a = *(const v2f*)(xrow + k4 * 4);
        v2f b = wfr[k4 * 32];
        acc = __builtin_amdgcn_wmma_f32_16x16x4_f32(
            /*neg_a=*/false, a, /*neg_b=*/false, b,
            /*c_mod=*/(short)0, acc, /*reuse_a=*/false, /*reuse_b=*/false);
    }

    float* ytile = Y + (size_t)(m0 + 8 * kh) * N + n0 + lm;
#pragma unroll
    for (int v = 0; v < 8; ++v)
        ytile[(size_t)v * N] = acc[v];
}

// ---------------- deterministic column sum / sumsq partials -----------------
__global__ void col_partial(const float* __restrict__ Y,
                            float* __restrict__ ps, float* __restrict__ pq,
                            int N, int rowsPerBlk) {
    int c = threadIdx.x;
    const float* p = Y + (size_t)blockIdx.x * rowsPerBlk * N + c;
    float s = 0.f, q = 0.f;
    for (int r = 0; r < rowsPerBlk; ++r) {
        float v = p[(size_t)r * N];
        s += v;
        q += v * v;
    }
    ps[(size_t)blockIdx.x * N + c] = s;
    pq[(size_t)blockIdx.x * N + c] = q;
}

// ---------------- fold BN into per-channel scale/shift ----------------------
__global__ void bn_coef(const float* __restrict__ ps, const float* __restrict__ pq,
                        const float* __restrict__ g, const float* __restrict__ b,
                        float* __restrict__ coef, int N, int nblk, float invM) {
    int c = blockIdx.x * blockDim.x + threadIdx.x;
    if (c >= N) return;
    float s = 0.f, q = 0.f;
    for (int i = 0; i < nblk; ++i) {
        s += ps[(size_t)i * N + c];
        q += pq[(size_t)i * N + c];
    }
    float m   = s * invM;
    float var = fmaxf(q * invM - m * m, 0.f);
    float a   = g[c] * rsqrtf(var + BN_EPS);
    coef[c]     = a;
    coef[N + c] = b[c] - m * a;
}

// ---------------- out0: data0 + upsample(relu(bn(Y10))), C = 96 -------------
__global__ void ew_out0(const float* __restrict__ data0, const float* __restrict__ Y10,
                        const float* __restrict__ coef, float* __restrict__ out, int nvec) {
    int idx = blockIdx.x * blockDim.x + threadIdx.x;
    if (idx >= nvec) return;
    int row = idx / 24;                 // C4 = 24
    int c   = (idx - row * 24) * 4;
    int par = row >> 3;
    float4 d = *(const float4*)(data0 + (size_t)row * 96 + c);
    float4 y = *(const float4*)(Y10 + (size_t)par * 96 + c);
    float4 o;
    o.x = d.x + fmaxf(0.f, coef[c + 0] * y.x + coef[96 + c + 0]);
    o.y = d.y + fmaxf(0.f, coef[c + 1] * y.y + coef[96 + c + 1]);
    o.z = d.z + fmaxf(0.f, coef[c + 2] * y.z + coef[96 + c + 2]);
    o.w = d.w + fmaxf(0.f, coef[c + 3] * y.w + coef[96 + c + 3]);
    *(float4*)(out + (size_t)row * 96 + c) = o;
}

// ---------------- out1: relu(bn(Y)) + D -------------------------------------
__global__ void ew_bn_add(const float* __restrict__ Y, const float* __restrict__ D,
                          const float* __restrict__ coef, float* __restrict__ out,
                          int C4, int nvec) {
    int idx = blockIdx.x * blockDim.x + threadIdx.x;
    if (idx >= nvec) return;
    int row = idx / C4;
    int c   = (idx - row * C4) * 4;
    int C   = C4 * 4;
    size_t off = (size_t)row * C + c;
    float4 y = *(const float4*)(Y + off);
    float4 d = *(const float4*)(D + off);
    float4 o;
    o.x = d.x + fmaxf(0.f, coef[c + 0] * y.x + coef[C + c + 0]);
    o.y = d.y + fmaxf(0.f, coef[c + 1] * y.y + coef[C + c + 1]);
    o.z = d.z + fmaxf(0.f, coef[c + 2] * y.z + coef[C + c + 2]);
    o.w = d.w + fmaxf(0.f, coef[c + 3] * y.w + coef[C + c + 3]);
    *(float4*)(out + off) = o;
}

// ---------------- out2: relu(bn(Ya)) + relu(bn(Yb)) -------------------------
__global__ void ew_bn2(const float* __restrict__ Ya, const float* __restrict__ Yb,
                       const float* __restrict__ ca, const float* __restrict__ cb,
                       float* __restrict__ out, int C4, int nvec) {
    int idx = blockIdx.x * blockDim.x + threadIdx.x;
    if (idx >= nvec) return;
    int row = idx / C4;
    int c   = (idx - row * C4) * 4;
    int C   = C4 * 4;
    size_t off = (size_t)row * C + c;
    float4 a = *(const float4*)(Ya + off);
    float4 b = *(const float4*)(Yb + off);
    float4 o;
    o.x = fmaxf(0.f, ca[c + 0] * a.x + ca[C + c + 0]) + fmaxf(0.f, cb[c + 0] * b.x + cb[C + c + 0]);
    o.y = fmaxf(0.f, ca[c + 1] * a.y + ca[C + c + 1]) + fmaxf(0.f, cb[c + 1] * b.y + cb[C + c + 1]);
    o.z = fmaxf(0.f, ca[c + 2] * a.z + ca[C + c + 2]) + fmaxf(0.f, cb[c + 2] * b.z + cb[C + c + 2]);
    o.w = fmaxf(0.f, ca[c + 3] * a.w + ca[C + c + 3]) + fmaxf(0.f, cb[c + 3] * b.w + cb[C + c + 3]);
    *(float4*)(out + off) = o;
}

// ---------------------------------------------------------------------------
extern "C" void kernel_launch(void* const* d_in, const int* in_sizes, int n_in,
                              void* d_out, int out_size, void* d_ws, size_t ws_size,
                              hipStream_t stream) {
    (void)in_sizes; (void)n_in; (void)out_size; (void)ws_size;

    const float* data0 = (const float*)d_in[0];
    const float* data1 = (const float*)d_in[1];
    const float* w01 = (const float*)d_in[2];
    const float* g01 = (const float*)d_in[3];
    const float* b01 = (const float*)d_in[4];
    const float* w02 = (const float*)d_in[5];
    const float* g02 = (const float*)d_in[6];
    const float* b02 = (const float*)d_in[7];
    const float* w10 = (const float*)d_in[8];
    const float* g10 = (const float*)d_in[9];
    const float* b10 = (const float*)d_in[10];
    const float* w12 = (const float*)d_in[11];
    const float* g12 = (const float*)d_in[12];
    const float* b12 = (const float*)d_in[13];

    float* out0 = (float*)d_out;                   // [N6,96]
    float* out1 = out0 + (size_t)N6 * 96;          // [N5,192]
    float* out2 = out1 + (size_t)N5 * 192;         // [N4,384]

    // ---- workspace layout (fp32) ----
    float* ws   = (float*)d_ws;
    float* p0   = ws;                                   // [N5,96]
    float* p00  = p0  + (size_t)N5 * 96;                // [N4,96]
    float* p1   = p00 + (size_t)N4 * 96;                // [N4,192]
    float* Y10  = p1  + (size_t)N4 * 192;               // [N5,96]
    float* Y01  = Y10 + (size_t)N5 * 96;                // [N5,192]
    float* Y02  = Y01 + (size_t)N5 * 192;               // [N4,384]
    float* Y12  = Y02 + (size_t)N4 * 384;               // [N4,384]
    float* ps10 = Y12 + (size_t)N4 * 384;               // [64,96]
    float* pq10 = ps10 + 64 * 96;
    float* ps01 = pq10 + 64 * 96;                       // [64,192]
    float* pq01 = ps01 + 64 * 192;
    float* ps02 = pq01 + 64 * 192;                      // [8,384]
    float* pq02 = ps02 + 8 * 384;
    float* ps12 = pq02 + 8 * 384;                       // [8,384]
    float* pq12 = ps12 + 8 * 384;
    float* c10  = pq12 + 8 * 384;                       // [2*96]
    float* c01  = c10 + 2 * 96;                         // [2*192]
    float* c02  = c01 + 2 * 192;                        // [2*384]
    float* c12  = c02 + 2 * 384;                        // [2*384]
    float* w10p = c12 + 2 * 384;                        // [192*96]  packed
    float* w01p = w10p + 192 * 96;                      // [96*192]  packed
    float* w02p = w01p + 96 * 192;                      // [96*384]  packed
    float* w12p = w02p + 96 * 384;                      // [192*384] packed

    const int TB = 256;

    // ---- pack weights into B-fragment order (tiny) ----
    pack_w_kernel<<<(192 * 96 / 2 + TB - 1) / TB, TB, 0, stream>>>(w10, w10p, 96, 192);
    pack_w_kernel<<<(96 * 192 / 2 + TB - 1) / TB, TB, 0, stream>>>(w01, w01p, 192, 96);
    pack_w_kernel<<<(96 * 384 / 2 + TB - 1) / TB, TB, 0, stream>>>(w02, w02p, 384, 96);
    pack_w_kernel<<<(192 * 384 / 2 + TB - 1) / TB, TB, 0, stream>>>(w12, w12p, 384, 192);

    // ---- pooling chain ----
    pool8_kernel<<<(N5 * 24 + TB - 1) / TB, TB, 0, stream>>>(data0, p0, 24, N5 * 24);
    pool8_kernel<<<(N4 * 24 + TB - 1) / TB, TB, 0, stream>>>(p0, p00, 24, N4 * 24);
    pool8_kernel<<<(N4 * 48 + TB - 1) / TB, TB, 0, stream>>>(data1, p1, 48, N4 * 48);

    // ---- GEMMs: 8 waves per block, one 16x16 tile per wave ----
    gemm_wmma_f32<96, 192><<<(N5 / 16) * (96 / 16) / 8, TB, 0, stream>>>(data1, w10p, Y10, N5);
    gemm_wmma_f32<192, 96><<<(N5 / 16) * (192 / 16) / 8, TB, 0, stream>>>(p0, w01p, Y01, N5);
    gemm_wmma_f32<384, 96><<<(N4 / 16) * (384 / 16) / 8, TB, 0, stream>>>(p00, w02p, Y02, N4);
    gemm_wmma_f32<384, 192><<<(N4 / 16) * (384 / 16) / 8, TB, 0, stream>>>(p1, w12p, Y12, N4);

    // ---- BN batch stats (deterministic two-stage column reduction) ----
    col_partial<<<64, 96,  0, stream>>>(Y10, ps10, pq10, 96, 1024);
    col_partial<<<64, 192, 0, stream>>>(Y01, ps01, pq01, 192, 1024);
    col_partial<<<8,  384, 0, stream>>>(Y02, ps02, pq02, 384, 1024);
    col_partial<<<8,  384, 0, stream>>>(Y12, ps12, pq12, 384, 1024);

    bn_coef<<<1, 96,  0, stream>>>(ps10, pq10, g10, b10, c10, 96, 64, 1.0f / N5);
    bn_coef<<<1, 192, 0, stream>>>(ps01, pq01, g01, b01, c01, 192, 64, 1.0f / N5);
    bn_coef<<<1, 384, 0, stream>>>(ps02, pq02, g02, b02, c02, 384, 8, 1.0f / N4);
    bn_coef<<<1, 384, 0, stream>>>(ps12, pq12, g12, b12, c12, 384, 8, 1.0f / N4);

    // ---- fused normalize + relu + residual outputs ----
    ew_out0<<<(N6 * 24 + TB - 1) / TB, TB, 0, stream>>>(data0, Y10, c10, out0, N6 * 24);
    ew_bn_add<<<(N5 * 48 + TB - 1) / TB, TB, 0, stream>>>(Y01, data1, c01, out1, 48, N5 * 48);
    ew_bn2<<<(N4 * 96 + TB - 1) / TB, TB, 0, stream>>>(Y02, Y12, c02, c12, out2, 96, N4 * 96);
}